// VolAttn3d_27109833573057
// MI455X (gfx1250) — compile-verified
//
#include <hip/hip_runtime.h>

typedef float v2f __attribute__((ext_vector_type(2)));
typedef float v8f __attribute__((ext_vector_type(8)));

#define NCTX 4
#define C16 16
#define WAVES_PER_BLOCK 8
#define TILES_PER_WAVE 8

__global__ __launch_bounds__(256) void volattn3d_wmma_kernel(
    const float* __restrict__ q,      // (C, V)
    const float* __restrict__ k,      // (N, C, V)
    const float* __restrict__ v,      // (N, C, V)
    const float* __restrict__ w_proj, // (C, C) row-major: w[o*16 + c]
    const float* __restrict__ b_proj, // (C)
    float* __restrict__ out,          // (C, V)
    int V)
{
    const int lane = threadIdx.x & 31;
    const int wave = threadIdx.x >> 5;
    const int half = lane >> 4;   // which K-pair of each 4-channel group this lane owns
    const int j    = lane & 15;   // voxel-in-tile; also A-matrix row M

    // Channel ownership chosen to match the WMMA f32 4x16 B layout directly:
    // lane owns channels c(r) = 4*(r>>1) + 2*half + (r&1), r = 0..7.
    // For K-chunk kk, B operand is exactly (x[2kk], x[2kk+1]). No shuffles.
    int chan[8];
#pragma unroll
    for (int r = 0; r < 8; ++r) chan[r] = 4 * (r >> 1) + 2 * half + (r & 1);

    // ---- A operand (w_proj) in V_WMMA_F32_16X16X4_F32 A layout, loaded once ----
    // lane: row M = j, chunk kk holds K = 4kk + 2*half + {0,1}
    v2f a[4];
#pragma unroll
    for (int kk = 0; kk < 4; ++kk) {
        const float* wp = w_proj + j * C16 + 4 * kk + 2 * half;
        a[kk].x = wp[0];
        a[kk].y = wp[1];
    }
    // bias preload in C/D layout: VGPR r -> out-channel r + 8*half
    float bias[8];
#pragma unroll
    for (int r = 0; r < 8; ++r) bias[r] = b_proj[r + 8 * half];

    const float scale = 0.25f; // 1/sqrt(C=16)
    const int tile0 = (blockIdx.x * WAVES_PER_BLOCK + wave) * TILES_PER_WAVE;

    for (int t = 0; t < TILES_PER_WAVE; ++t) {
        const int vbase = (tile0 + t) * 16;
        if (vbase + 16 > V) break;          // wave-uniform guard: EXEC stays full
        const int pos = vbase + j;          // this lane's voxel

        // ---- stream q (this lane's 8 owned channels), non-temporal ----
        float qv[8];
#pragma unroll
        for (int r = 0; r < 8; ++r)
            qv[r] = __builtin_nontemporal_load(q + chan[r] * V + pos);

        // ---- scores: dot over 16 channels = 8 local + 8 from partner half ----
        float s[NCTX];
#pragma unroll
        for (int n = 0; n < NCTX; ++n) {
            float part = 0.0f;
#pragma unroll
            for (int r = 0; r < 8; ++r) {
                float kv = __builtin_nontemporal_load(
                    k + (n * C16 + chan[r]) * V + pos);
                part = fmaf(qv[r], kv, part);
            }
            part += __shfl_xor(part, 16, 32);   // add partner half's channels
            s[n] = part * scale;
        }

        // ---- softmax over N=4 (registers) ----
        float m = fmaxf(fmaxf(s[0], s[1]), fmaxf(s[2], s[3]));
        float e[NCTX];
        float sum = 0.0f;
#pragma unroll
        for (int n = 0; n < NCTX; ++n) { e[n] = __expf(s[n] - m); sum += e[n]; }
        float inv = __builtin_amdgcn_rcpf(sum);
#pragma unroll
        for (int n = 0; n < NCTX; ++n) e[n] *= inv;

        // ---- x = attn . v, held directly in WMMA B-operand channel order ----
        float x[8];
#pragma unroll
        for (int r = 0; r < 8; ++r) x[r] = 0.0f;
#pragma unroll
        for (int n = 0; n < NCTX; ++n) {
#pragma unroll
            for (int r = 0; r < 8; ++r) {
                float vv = __builtin_nontemporal_load(
                    v + (n * C16 + chan[r]) * V + pos);
                x[r] = fmaf(e[n], vv, x[r]);
            }
        }

        // ---- 4 chained V_WMMA_F32_16X16X4_F32: out = W(16x16) * x(16x16) + b ----
        v8f acc;
#pragma unroll
        for (int r = 0; r < 8; ++r) acc[r] = bias[r];
#pragma unroll
        for (int kk = 0; kk < 4; ++kk) {
            v2f b;                       // B layout: lane holds K = 4kk+2h+{0,1}
            b.x = x[2 * kk + 0];
            b.y = x[2 * kk + 1];
            acc = __builtin_amdgcn_wmma_f32_16x16x4_f32(
                false, a[kk], false, b, (short)0, acc, false, false);
        }

        // ---- store: C/D layout -> out-channel r+8*half, voxel j ----
#pragma unroll
        for (int r = 0; r < 8; ++r)
            __builtin_nontemporal_store(acc[r], out + (r + 8 * half) * V + pos);
    }
}

extern "C" void kernel_launch(void* const* d_in, const int* in_sizes, int n_in,
                              void* d_out, int out_size, void* d_ws, size_t ws_size,
                              hipStream_t stream) {
    const float* q      = (const float*)d_in[0];
    const float* k      = (const float*)d_in[1];
    const float* v      = (const float*)d_in[2];
    const float* w_proj = (const float*)d_in[3];
    const float* b_proj = (const float*)d_in[4];
    float* out = (float*)d_out;

    const int V = in_sizes[0] / C16;                        // D*H*W = 1,048,576
    const int voxPerBlock = WAVES_PER_BLOCK * TILES_PER_WAVE * 16; // 1024
    const int blocks = (V + voxPerBlock - 1) / voxPerBlock; // 1024

    volattn3d_wmma_kernel<<<blocks, 32 * WAVES_PER_BLOCK, 0, stream>>>(
        q, k, v, w_proj, b_proj, out, V);
}